// BahdanauAttnDecoderRNN_36558761624530
// MI455X (gfx1250) — compile-verified
//
#include <hip/hip_runtime.h>
#include <hip/hip_bf16.h>
#include <math.h>

// ---------------------------------------------------------------------------
// BahdanauAttnDecoderRNN single step, H=1024, V=32000, S=2048, batch=1.
// Dead-code: softmax over scalar == 1.0  =>  attn_weights = ones(S),
// attn_applied = column-sum of encoder_outputs, attn_W/attn_b unused.
// All GEMVs run through V_WMMA_F32_16X16X4_F32 (fp32 matrix path, wave32),
// with b128 weight/vector loads feeding two WMMAs per iteration.
// ---------------------------------------------------------------------------

#define H 1024
#define V 32000
#define S 2048

typedef float v2f __attribute__((ext_vector_type(2)));
typedef float v4f __attribute__((ext_vector_type(4)));
typedef float v8f __attribute__((ext_vector_type(8)));

// ---------------------------------------------------------------------------
// K1: partial column sums of encoder_outputs. 16 blocks x 256 threads,
// block b reduces rows [b*128, (b+1)*128) into partial[b*H + j].
// ---------------------------------------------------------------------------
__global__ void k_colsum(const float* __restrict__ E, float* __restrict__ partial) {
    const int b = blockIdx.x;
    const int t = threadIdx.x;
    const float* base = E + (size_t)b * 128 * H;
    for (int j = t; j < H; j += 256) {
        float acc = 0.0f;
        const float* p = base + j;
        #pragma unroll 4
        for (int s = 0; s < 128; ++s) acc += p[(size_t)s * H];
        partial[b * H + j] = acc;
    }
}

// ---------------------------------------------------------------------------
// K2: rnn_input = relu(concat(embedding[word], colsum)), attn_weights = 1.0
// 4 blocks x 256 threads (j = 0..1023).
// ---------------------------------------------------------------------------
__global__ void k_build_input(const float* __restrict__ partial,
                              const float* __restrict__ embedding,
                              const int* __restrict__ word,
                              float* __restrict__ rnn_input,
                              float* __restrict__ attn_out) {
    const int j = blockIdx.x * blockDim.x + threadIdx.x;
    if (j < H) {
        float s = 0.0f;
        #pragma unroll
        for (int b = 0; b < 16; ++b) s += partial[b * H + j];
        rnn_input[H + j] = fmaxf(s, 0.0f);
        const int w = word[0];
        rnn_input[j] = fmaxf(embedding[(size_t)w * H + j], 0.0f);
    }
    const int total = gridDim.x * blockDim.x; // 1024
    for (int i = j; i < S; i += total) attn_out[i] = 1.0f;
}

// ---------------------------------------------------------------------------
// K3: GEMV out[m] = sum_k W[m,k]*vec[k] + bias[m] via WMMA fp32 16x16x4.
// One wave per 16-row tile; K consumed 8-at-a-time:
//   lanes 0-15  load W[m, k   .. k+3 ]  (A slots {0,1} of two WMMAs)
//   lanes 16-31 load W[m, k+4 .. k+7 ]  (A slots {2,3} of two WMMAs)
//   B column 0 carries the matching vec values (lanes 0 and 16 only).
// D column 0: lane 0 -> M=0..7 in c[0..7], lane 16 -> M=8..15.
// blockDim = 128 (4 waves). Grids sized exactly (M % 16 == 0, K % 16 == 0).
// ---------------------------------------------------------------------------
__global__ void k_gemv_wmma(const float* __restrict__ W,
                            const float* __restrict__ vec,
                            const float* __restrict__ bias,
                            float* __restrict__ out,
                            int M, int K) {
    extern __shared__ float sv[];               // K floats: vector staged in LDS
    const int tid = threadIdx.x;
    for (int i = tid; i < K; i += blockDim.x) sv[i] = vec[i];
    __syncthreads();

    const int lane  = tid & 31;
    const int wave  = tid >> 5;
    const int tile  = blockIdx.x * (blockDim.x >> 5) + wave;
    const int rowBase = tile * 16;
    if (rowBase >= M) return;                   // wave-uniform exit

    const int m    = lane & 15;
    const int qoff = (lane >> 4) * 4;           // 0 (lanes 0-15) or 4 (lanes 16-31)
    const float* wrow = W + (size_t)(rowBase + m) * K + qoff;
    const bool keep = (lane == 0) || (lane == 16);

    v8f c = {0.f, 0.f, 0.f, 0.f, 0.f, 0.f, 0.f, 0.f};
    #pragma unroll 2
    for (int k = 0; k < K; k += 8) {
        __builtin_prefetch(wrow + k + 512, 0, 0);   // global_prefetch_b8, ~2KB ahead
        v4f w = *(const v4f*)(wrow + k);            // global_load_b128 (all bytes used)
        v4f t = *(const v4f*)(sv + k + qoff);       // ds_load_b128
        v2f a0 = {w.x, w.y};
        v2f a1 = {w.z, w.w};
        v2f b0, b1;
        b0.x = keep ? t.x : 0.0f;  b0.y = keep ? t.y : 0.0f;
        b1.x = keep ? t.z : 0.0f;  b1.y = keep ? t.w : 0.0f;
        c = __builtin_amdgcn_wmma_f32_16x16x4_f32(false, a0, false, b0,
                                                  (short)0, c, false, false);
        c = __builtin_amdgcn_wmma_f32_16x16x4_f32(false, a1, false, b1,
                                                  (short)0, c, false, false);
    }

    if (keep) {
        const int r0 = rowBase + ((lane >> 4) << 3);  // lane0 -> +0, lane16 -> +8
        v4f lo, hi;
        #pragma unroll
        for (int r = 0; r < 4; ++r) {
            lo[r] = c[r]     + bias[r0 + r];
            hi[r] = c[4 + r] + bias[r0 + 4 + r];
        }
        *(v4f*)(out + r0)     = lo;               // global_store_b128
        *(v4f*)(out + r0 + 4) = hi;               // global_store_b128
    }
}

// ---------------------------------------------------------------------------
// K4: fused GRU gates (PyTorch layout [r, z, n]). 4 blocks x 256 threads.
// ---------------------------------------------------------------------------
__global__ void k_gru_gate(const float* __restrict__ gi, const float* __restrict__ gh,
                           const float* __restrict__ h, float* __restrict__ h_new) {
    const int j = blockIdx.x * blockDim.x + threadIdx.x;
    if (j >= H) return;
    const float r = 1.0f / (1.0f + expf(-(gi[j]         + gh[j])));
    const float z = 1.0f / (1.0f + expf(-(gi[H + j]     + gh[H + j])));
    const float n = tanhf(gi[2 * H + j] + r * gh[2 * H + j]);
    h_new[j] = (1.0f - z) * n + z * h[j];
}

// ---------------------------------------------------------------------------
// K5: log_softmax over V=32000 logits. Single block, 1024 threads,
// deterministic fixed-order tree reductions in LDS.
// ---------------------------------------------------------------------------
__global__ void k_logsoftmax(const float* __restrict__ logits, float* __restrict__ out) {
    __shared__ float red[1024];
    const int t = threadIdx.x;

    float m = -INFINITY;
    for (int i = t; i < V; i += 1024) m = fmaxf(m, logits[i]);
    red[t] = m;
    __syncthreads();
    for (int s = 512; s > 0; s >>= 1) {
        if (t < s) red[t] = fmaxf(red[t], red[t + s]);
        __syncthreads();
    }
    const float gmax = red[0];
    __syncthreads();

    float sum = 0.0f;
    for (int i = t; i < V; i += 1024) sum += expf(logits[i] - gmax);
    red[t] = sum;
    __syncthreads();
    for (int s = 512; s > 0; s >>= 1) {
        if (t < s) red[t] += red[t + s];
        __syncthreads();
    }
    const float lse = gmax + logf(red[0]);

    for (int i = t; i < V; i += 1024) out[i] = logits[i] - lse;
}

// ---------------------------------------------------------------------------
// Host-side orchestration.
// Input order: 0 word_input(int,1) 1 last_hidden(H) 2 encoder_outputs(S*H)
//   3 embedding(V*H) 4 attn_W(unused) 5 attn_b(unused) 6 gru_W_ih(3H*2H)
//   7 gru_W_hh(3H*H) 8 gru_b_ih(3H) 9 gru_b_hh(3H) 10 out_W(V*H) 11 out_b(V)
// Output: [0,V) log_softmax | [V, V+H) h_new | [V+H, V+H+S) attn_weights
// ---------------------------------------------------------------------------
extern "C" void kernel_launch(void* const* d_in, const int* in_sizes, int n_in,
                              void* d_out, int out_size, void* d_ws, size_t ws_size,
                              hipStream_t stream) {
    const int*   word      = (const int*)  d_in[0];
    const float* last_h    = (const float*)d_in[1];
    const float* enc       = (const float*)d_in[2];
    const float* embedding = (const float*)d_in[3];
    const float* W_ih      = (const float*)d_in[6];
    const float* W_hh      = (const float*)d_in[7];
    const float* b_ih      = (const float*)d_in[8];
    const float* b_hh      = (const float*)d_in[9];
    const float* out_W     = (const float*)d_in[10];
    const float* out_b     = (const float*)d_in[11];

    float* out      = (float*)d_out;
    float* out_h    = out + V;          // h_new region
    float* out_attn = out + V + H;      // attn_weights region

    float* ws        = (float*)d_ws;
    float* rnn_input = ws;              // 2H   = 2048
    float* gi        = ws + 2048;       // 3H   = 3072
    float* gh        = ws + 5120;       // 3H   = 3072
    float* logits    = ws + 8192;       // V    = 32000
    float* partial   = ws + 40192;      // 16*H = 16384  (total ~226 KB)

    // 1) column sums of encoder_outputs
    k_colsum<<<16, 256, 0, stream>>>(enc, partial);
    // 2) rnn_input = relu(concat(embed, colsum)); attn_weights = 1.0
    k_build_input<<<4, 256, 0, stream>>>(partial, embedding, word, rnn_input, out_attn);
    // 3) gi = W_ih @ rnn_input + b_ih   (3072 x 2048) : 192 tiles / 4 waves
    k_gemv_wmma<<<48, 128, 2 * H * sizeof(float), stream>>>(W_ih, rnn_input, b_ih, gi,
                                                            3 * H, 2 * H);
    // 4) gh = W_hh @ h + b_hh           (3072 x 1024)
    k_gemv_wmma<<<48, 128, H * sizeof(float), stream>>>(W_hh, last_h, b_hh, gh,
                                                        3 * H, H);
    // 5) GRU gate fusion -> h_new (written straight into d_out)
    k_gru_gate<<<4, 256, 0, stream>>>(gi, gh, last_h, out_h);
    // 6) logits = out_W @ h_new + out_b (32000 x 1024) : 2000 tiles / 4 waves
    k_gemv_wmma<<<500, 128, H * sizeof(float), stream>>>(out_W, out_h, out_b, logits,
                                                         V, H);
    // 7) log_softmax -> d_out[0..V)
    k_logsoftmax<<<1, 1024, 0, stream>>>(logits, out);
}